// MultiHeadSelfAttentionWithRoPE_45810121179348
// MI455X (gfx1250) — compile-verified
//
#include <hip/hip_runtime.h>
#include <math.h>

typedef float v2f __attribute__((ext_vector_type(2)));
typedef float v8f __attribute__((ext_vector_type(8)));
typedef unsigned int u32x4 __attribute__((ext_vector_type(4)));
typedef int i32x4 __attribute__((ext_vector_type(4)));
typedef int i32x8 __attribute__((ext_vector_type(8)));

#define S_LEN   4096
#define D_MODEL 768
#define N_HEADS 12
#define HEAD_DIM 64
#define LDK 36   // 32 data + 4 pad dwords per row (TDM pad_interval=enc4, pad_amount=enc3)
#define LQ  68   // 64 data + 4 pad dwords per row (TDM pad_interval=enc5, pad_amount=enc3)

__device__ __forceinline__ v8f wmma_f32(v2f a, v2f b, v8f c) {
  // D(16x16) = A(16x4) * B(4x16) + C, fp32
  return __builtin_amdgcn_wmma_f32_16x16x4_f32(false, a, false, b, (short)0, c,
                                               false, false);
}

// ---------------------------------------------------------------------------
// Tensor Data Mover: DMA a 2-D fp32 tile (tile_w x tile_h elems, row stride
// stride_elems) from global memory into LDS, with hardware row padding.
// D# bit layout per CDNA5 ISA 8.3/8.4. Issued by one wave; completion via
// TENSORcnt (s_wait_tensorcnt) then a workgroup barrier.
// ---------------------------------------------------------------------------
__device__ __forceinline__ void tdm_load_2d(void* lds_dst, const float* src,
                                            int tile_w, int tile_h,
                                            int stride_elems, int rows_total,
                                            int pad_int_enc, int pad_amt_enc) {
  const unsigned long long ga = (unsigned long long)(uintptr_t)src;
  u32x4 g0;
  g0.x = 1u;                                   // count=1, user descriptor
  g0.y = (unsigned int)(uintptr_t)lds_dst;     // lds_addr (byte offset, low 32b)
  g0.z = (unsigned int)ga;                     // global_addr[31:0]
  g0.w = (unsigned int)(ga >> 32) | (2u << 30);// global_addr[56:32] | type=2
  i32x8 g1;
  g1[0] = (2 << 16)                  // data_size = 4B
        | (1 << 20)                  // pad_enable
        | (pad_int_enc << 22)        // pad every 2^(enc+1) dwords
        | (pad_amt_enc << 25);       // pad (enc+1) dwords
  g1[1] = (stride_elems & 0xFFFF) << 16;                       // tensor_dim0 lo
  g1[2] = (stride_elems >> 16) | ((rows_total & 0xFFFF) << 16);// dim0 hi|dim1 lo
  g1[3] = (rows_total >> 16) | (tile_w << 16);                 // dim1 hi|tile_dim0
  g1[4] = tile_h;                                              // tile_dim1 (tile_dim2=0)
  g1[5] = stride_elems;                                        // tensor_dim0_stride lo
  g1[6] = 0;
  g1[7] = 0;
  const i32x4 gz4 = {0, 0, 0, 0};
  const i32x8 gz8 = {0, 0, 0, 0, 0, 0, 0, 0};
  __builtin_amdgcn_tensor_load_to_lds(g0, g1, gz4, gz4, gz8, 0);
}

// ---------------------------------------------------------------------------
// GEMM: Out[m,n] = sum_d A[m,d] * W[n,d]  (A @ W^T), 128x64 block tile,
// 8 waves each computing 32x32 (4 accumulators), TDM double-buffered K loop.
// Optional fused RoPE epilogue (Q/K projections).
// ---------------------------------------------------------------------------
__global__ __launch_bounds__(256) void gemm_rope_kernel(
    const float* __restrict__ A,   // [4096 x 768]
    const float* __restrict__ W,   // [768 x 768]
    float* __restrict__ Out,       // [4096 x 768]
    const int* __restrict__ tokpos,
    int apply_rope)
{
  __shared__ float Xs [2][128 * LDK];
  __shared__ float Wls[2][ 64 * LDK];

  const int m0   = blockIdx.x * 128;
  const int n0   = blockIdx.y * 64;
  const int tid  = threadIdx.x;
  const int wid  = tid >> 5;
  const int lane = tid & 31;
  const int hl   = lane >> 4;        // K-pair half for f32 WMMA operands
  const int lo   = lane & 15;
  const int wm   = (wid >> 1) * 32;  // 0,32,64,96
  const int wn   = (wid & 1) * 32;   // 0,32

  v8f c00 = {}, c01 = {}, c10 = {}, c11 = {};

  if (wid == 0) {   // prologue: DMA first K-slab into buffer 0
    tdm_load_2d(&Xs [0][0], &A[(size_t)m0 * D_MODEL], 32, 128, D_MODEL, S_LEN,   4, 3);
    tdm_load_2d(&Wls[0][0], &W[(size_t)n0 * D_MODEL], 32,  64, D_MODEL, D_MODEL, 4, 3);
  }

  for (int it = 0; it < D_MODEL / 32; ++it) {
    const int kt  = it * 32;
    const int cur = it & 1;
    if (wid == 0) {
      if (it + 1 < D_MODEL / 32) {   // prefetch next slab into other buffer
        tdm_load_2d(&Xs [cur ^ 1][0], &A[(size_t)m0 * D_MODEL + kt + 32], 32, 128, D_MODEL, S_LEN,   4, 3);
        tdm_load_2d(&Wls[cur ^ 1][0], &W[(size_t)n0 * D_MODEL + kt + 32], 32,  64, D_MODEL, D_MODEL, 4, 3);
        __builtin_amdgcn_s_wait_tensorcnt(2);  // in-order: current slab done
      } else {
        __builtin_amdgcn_s_wait_tensorcnt(0);
      }
    }
    __syncthreads();

    const float* xs = &Xs [cur][0];
    const float* ws = &Wls[cur][0];
    #pragma unroll
    for (int k0 = 0; k0 < 32; k0 += 4) {
      const int kk = k0 + 2 * hl;    // VGPR0: K={0|2}, VGPR1: K={1|3}
      v2f a0 = *(const v2f*)&xs[(wm +      lo) * LDK + kk];
      v2f a1 = *(const v2f*)&xs[(wm + 16 + lo) * LDK + kk];
      v2f b0 = *(const v2f*)&ws[(wn +      lo) * LDK + kk];
      v2f b1 = *(const v2f*)&ws[(wn + 16 + lo) * LDK + kk];
      c00 = wmma_f32(a0, b0, c00);
      c01 = wmma_f32(a0, b1, c01);
      c10 = wmma_f32(a1, b0, c10);
      c11 = wmma_f32(a1, b1, c11);
    }
    __syncthreads();
  }

  // Epilogue. C layout: lane lo = col, vgpr j = row (j + 8*hl).
  v8f accs[2][2] = {{c00, c01}, {c10, c11}};
  #pragma unroll
  for (int am = 0; am < 2; ++am) {
    #pragma unroll
    for (int an = 0; an < 2; ++an) {
      const v8f cc  = accs[am][an];
      const int col = n0 + wn + an * 16 + lo;
      if (apply_rope) {
        const float inv_freq =
            __expf(-0.28782313662425572f * (float)(((wn + an * 16 + lo) & 63) >> 1));
        const bool odd = (lane & 1) != 0;
        #pragma unroll
        for (int j = 0; j < 8; ++j) {
          const int row  = m0 + wm + am * 16 + j + 8 * hl;
          const float v  = cc[j];
          const float vp = __shfl_xor(v, 1, 32);
          float sn, cs;
          __sincosf((float)tokpos[row] * inv_freq, &sn, &cs);
          Out[(size_t)row * D_MODEL + col] = odd ? (vp * sn + v * cs) : (v * cs - vp * sn);
        }
      } else {
        #pragma unroll
        for (int j = 0; j < 8; ++j)
          Out[(size_t)(m0 + wm + am * 16 + j + 8 * hl) * D_MODEL + col] = cc[j];
      }
    }
  }
}

// ---------------------------------------------------------------------------
// Flash attention, causal. Block = (head, 128 query rows), 8 waves x 16 q.
// S^T = K*Q^T keeps softmax stats lane-local; O^T = V^T*P^T.
// Q tile + double-buffered K/V tiles arrive via TDM.
// ---------------------------------------------------------------------------
__global__ __launch_bounds__(256) void attn_fa_kernel(
    const float* __restrict__ Q, const float* __restrict__ K,
    const float* __restrict__ V, float* __restrict__ O)
{
  __shared__ float Qs[128 * LQ];
  __shared__ float Ks[2][16 * LQ];
  __shared__ float Vs[2][16 * LQ];

  const int tid  = threadIdx.x;
  const int wid  = tid >> 5;
  const int lane = tid & 31;
  const int hl   = lane >> 4;
  const int lo   = lane & 15;
  const int qb   = blockIdx.x;          // 0..31
  const int head = blockIdx.y;          // 0..11
  const int q0g  = qb * 128;
  const int cb   = head * HEAD_DIM;

  const int ntiles = qb * 8 + 8;        // causal kv coverage for this block

  if (wid == 0) {                       // DMA Q tile + first K/V tiles
    tdm_load_2d(&Qs[0],    &Q[(size_t)q0g * D_MODEL + cb], 64, 128, D_MODEL, S_LEN, 5, 3);
    tdm_load_2d(&Ks[0][0], &K[cb], 64, 16, D_MODEL, S_LEN, 5, 3);
    tdm_load_2d(&Vs[0][0], &V[cb], 64, 16, D_MODEL, S_LEN, 5, 3);
  }

  const int q0l = wid * 16;
  const int qg  = q0g + q0l + lo;       // this lane's query index
  v8f acc[4] = {};                      // O^T: d = t*16 + j + 8*hl, lane = query
  float m_i = -1e30f, l_i = 0.0f;

  for (int it = 0; it < ntiles; ++it) {
    const int kv0 = it * 16;
    const int cur = it & 1;
    if (wid == 0) {
      if (it + 1 < ntiles) {
        tdm_load_2d(&Ks[cur ^ 1][0], &K[(size_t)(kv0 + 16) * D_MODEL + cb], 64, 16, D_MODEL, S_LEN, 5, 3);
        tdm_load_2d(&Vs[cur ^ 1][0], &V[(size_t)(kv0 + 16) * D_MODEL + cb], 64, 16, D_MODEL, S_LEN, 5, 3);
        __builtin_amdgcn_s_wait_tensorcnt(2);
      } else {
        __builtin_amdgcn_s_wait_tensorcnt(0);
      }
    }
    __syncthreads();

    if (kv0 <= q0g + q0l + 15) {        // wave-uniform: EXEC all-ones for WMMA
      const float* ks = &Ks[cur][0];
      const float* vs = &Vs[cur][0];

      // S^T(16kv x 16q) = K_tile * Q_tile^T over d=64
      v8f st = {};
      #pragma unroll
      for (int d0 = 0; d0 < 64; d0 += 4) {
        const int dd = d0 + 2 * hl;
        v2f a = *(const v2f*)&ks[lo * LQ + dd];              // A[kv][d]
        v2f b = *(const v2f*)&Qs[(q0l + lo) * LQ + dd];      // B[d][q] = Q[q][d]
        st = wmma_f32(a, b, st);
      }

      // scale + causal mask + online softmax (kv axis in VGPRs: lane-local)
      float sa[8];
      float mnew = m_i;
      #pragma unroll
      for (int j = 0; j < 8; ++j) {
        const int kv = kv0 + j + 8 * hl;
        float s = st[j] * 0.125f;
        s = (kv <= qg) ? s : -1e30f;
        sa[j] = s;
        mnew = fmaxf(mnew, s);
      }
      const float alpha = __expf(m_i - mnew);
      float psum = 0.0f;
      #pragma unroll
      for (int j = 0; j < 8; ++j) {
        const float p = __expf(sa[j] - mnew);
        sa[j] = p;
        psum += p;
      }
      l_i = l_i * alpha + psum;
      m_i = mnew;
      #pragma unroll
      for (int t = 0; t < 4; ++t)
        #pragma unroll
        for (int j = 0; j < 8; ++j) acc[t][j] *= alpha;

      // half-swapped P^T copy for B-operand assembly
      float sw[8];
      #pragma unroll
      for (int j = 0; j < 8; ++j) sw[j] = __shfl_xor(sa[j], 16, 32);

      // O^T(d x q) += V^T(d x kv) * P^T(kv x q)
      #pragma unroll
      for (int t = 0; t < 4; ++t) {
        #pragma unroll
        for (int k0 = 0; k0 < 16; k0 += 4) {
          const int r0 = k0 + 2 * hl;
          v2f a;
          a.x = vs[r0 * LQ + t * 16 + lo];
          a.y = vs[(r0 + 1) * LQ + t * 16 + lo];
          const int  kc   = k0 & 7;
          const bool self = ((k0 >> 3) == hl);
          const float s0 = hl ? sa[kc + 2] : sa[kc];
          const float s1 = hl ? sa[kc + 3] : sa[kc + 1];
          const float w0 = hl ? sw[kc + 2] : sw[kc];
          const float w1 = hl ? sw[kc + 3] : sw[kc + 1];
          v2f b;
          b.x = self ? s0 : w0;
          b.y = self ? s1 : w1;
          acc[t] = wmma_f32(a, b, acc[t]);
        }
      }
    }
    __syncthreads();
  }

  const float inv_l = 1.0f / l_i;
  const int row = q0g + q0l + lo;
  #pragma unroll
  for (int t = 0; t < 4; ++t)
    #pragma unroll
    for (int j = 0; j < 8; ++j)
      O[(size_t)row * D_MODEL + cb + t * 16 + j + 8 * hl] = acc[t][j] * inv_l;
}

// ---------------------------------------------------------------------------
extern "C" void kernel_launch(void* const* d_in, const int* in_sizes, int n_in,
                              void* d_out, int out_size, void* d_ws, size_t ws_size,
                              hipStream_t stream) {
  (void)in_sizes; (void)n_in; (void)out_size; (void)ws_size;
  const float* x      = (const float*)d_in[0];
  const float* wq     = (const float*)d_in[1];
  const float* wk     = (const float*)d_in[2];
  const float* wv     = (const float*)d_in[3];
  const float* wo     = (const float*)d_in[4];
  const int*   tokpos = (const int*)d_in[5];
  float* out = (float*)d_out;

  const size_t plane = (size_t)S_LEN * D_MODEL;
  float* qws = (float*)d_ws;
  float* kws = qws + plane;
  float* vws = kws + plane;
  float* ows = vws + plane;

  dim3 blk(256);
  dim3 gproj(S_LEN / 128, D_MODEL / 64);   // 32 x 12
  gemm_rope_kernel<<<gproj, blk, 0, stream>>>(x, wq, qws, tokpos, 1);
  gemm_rope_kernel<<<gproj, blk, 0, stream>>>(x, wk, kws, tokpos, 1);
  gemm_rope_kernel<<<gproj, blk, 0, stream>>>(x, wv, vws, tokpos, 0);
  attn_fa_kernel<<<dim3(S_LEN / 128, N_HEADS), blk, 0, stream>>>(qws, kws, vws, ows);
  gemm_rope_kernel<<<gproj, blk, 0, stream>>>(ows, wo, out, tokpos, 0);
}